// GCN_4_44659069943894
// MI455X (gfx1250) — compile-verified
//
#include <hip/hip_runtime.h>
#include <hip/hip_bf16.h>

typedef float v2f __attribute__((ext_vector_type(2)));
typedef float v8f __attribute__((ext_vector_type(8)));

// ---------------- graph preprocessing ----------------

// cnt=0, deg=1.0 (self loop weight)
__global__ void k_init(int* cnt, float* deg, int n) {
  int i = blockIdx.x * blockDim.x + threadIdx.x;
  if (i < n) { cnt[i] = 0; deg[i] = 1.0f; }
}

// per-edge: in-degree count (int) and weighted degree (float)
__global__ void k_count(const int* __restrict__ ei, const float* __restrict__ ew,
                        int* cnt, float* deg, int E) {
  int e = blockIdx.x * blockDim.x + threadIdx.x;
  if (e < E) {
    int d = ei[E + e];               // dst row of edge_index
    atomicAdd(&cnt[d], 1);
    atomicAdd(&deg[d], ew[e]);
  }
}

// deg -> dis = rsqrt(deg) in place (deg >= 1 always); nself = dis^2
__global__ void k_dis(float* deg_dis, float* nself, int n) {
  int i = blockIdx.x * blockDim.x + threadIdx.x;
  if (i < n) {
    float v = rsqrtf(deg_dis[i]);
    deg_dis[i] = v;
    nself[i] = v * v;
  }
}

// exclusive scan, 512 elems per block
__global__ void k_scan1(const int* __restrict__ cnt, int* rowp, int* bsum, int n) {
  __shared__ int s[512];
  int i = blockIdx.x * 512 + threadIdx.x;
  int val = (i < n) ? cnt[i] : 0;
  s[threadIdx.x] = val;
  __syncthreads();
  for (int off = 1; off < 512; off <<= 1) {
    int t = (threadIdx.x >= off) ? s[threadIdx.x - off] : 0;
    __syncthreads();
    s[threadIdx.x] += t;
    __syncthreads();
  }
  if (i < n) rowp[i] = s[threadIdx.x] - val;     // exclusive
  if (threadIdx.x == 511) bsum[blockIdx.x] = s[511];
}

__global__ void k_scan2(const int* __restrict__ bsum, int* boff, int nb) {
  __shared__ int s[512];
  int val = (threadIdx.x < nb) ? bsum[threadIdx.x] : 0;
  s[threadIdx.x] = val;
  __syncthreads();
  for (int off = 1; off < 512; off <<= 1) {
    int t = (threadIdx.x >= off) ? s[threadIdx.x - off] : 0;
    __syncthreads();
    s[threadIdx.x] += t;
    __syncthreads();
  }
  if (threadIdx.x < nb) boff[threadIdx.x] = s[threadIdx.x] - val;
}

__global__ void k_scan3(int* rowp, const int* __restrict__ boff, int* cur, int n) {
  int i = blockIdx.x * blockDim.x + threadIdx.x;
  if (i < n) {
    int r = rowp[i] + boff[i >> 9];
    rowp[i] = r;
    cur[i] = r;                      // scatter cursor; ends up == row end
  }
}

// scatter edges into CSR-by-dst, norm computed inline
__global__ void k_scatter(const int* __restrict__ ei, const float* __restrict__ ew,
                          const float* __restrict__ dis, int* cur,
                          int* csrS, float* csrN, int E) {
  int e = blockIdx.x * blockDim.x + threadIdx.x;
  if (e < E) {
    int s = ei[e];
    int d = ei[E + e];
    float nv = dis[s] * ew[e] * dis[d];
    int pos = atomicAdd(&cur[d], 1);
    csrS[pos] = s;
    csrN[pos] = nv;
  }
}

// ---------------- layer compute ----------------

// layer 1 GEMM: t[N,128] = x[N,1] @ W1[1,128] (rank-1)
__global__ void k_gemm1(const float* __restrict__ x, const float* __restrict__ W1,
                        float* __restrict__ t, int n) {
  int idx = blockIdx.x * blockDim.x + threadIdx.x;   // n*32 float4 slots
  if (idx < n * 32) {
    int v = idx >> 5, q = idx & 31;
    float xv = x[v];
    float4 w = ((const float4*)W1)[q];
    float4 r;
    r.x = xv * w.x; r.y = xv * w.y; r.z = xv * w.z; r.w = xv * w.w;
    ((float4*)t)[idx] = r;
  }
}

// f32 WMMA GEMM: T[n,fo] = A[n,fi] @ W[fi,fo]; one wave per 32x32 tile
// (2x2 register tiling of 16x16 WMMA tiles -> W traffic /2, A traffic /2).
// All operand streams use incrementing pointers so the backend can fold the
// per-k addressing into immediate-offset loads on a single stepped base.
// fi % 4 == 0, fo % 32 == 0, n % 32 == 0.
__global__ __launch_bounds__(256)
void k_gemm(const float* __restrict__ A, const float* __restrict__ W,
            float* __restrict__ T, int n, int fi, int fo) {
  int wave = threadIdx.x >> 5;
  int lane = threadIdx.x & 31;
  int rowBase = (blockIdx.x * 8 + wave) * 32;
  int colBase = blockIdx.y * 32;
  if (rowBase >= n) return;                 // wave-uniform: EXEC stays all-ones
  int lr  = lane & 15;
  int hi  = lane >> 4;                      // 0: K,K+1  1: K+2,K+3 (ISA 16x4 f32 A layout)
  int col0 = colBase + lr;
  int col1 = col0 + 16;
  const float* a0p = A + (rowBase + lr) * fi + hi * 2;        // 8B-aligned per k
  const float* a1p = a0p + 16 * fi;
#if __has_builtin(__builtin_amdgcn_wmma_f32_16x16x4_f32)
  const float* wp = W + (hi * 2) * fo + col0;   // rows kb/kb+1, cols col0/col0+16
  const int wstep = 4 * fo;
  v8f c00 = {}, c01 = {}, c10 = {}, c11 = {};
  for (int k = 0; k < fi; k += 4) {
    v2f a0 = *(const v2f*)a0p;
    v2f a1 = *(const v2f*)a1p;
    a0p += 4;
    a1p += 4;
    v2f b0, b1;
    b0.x = wp[0];   b0.y = wp[fo];
    b1.x = wp[16];  b1.y = wp[fo + 16];
    wp += wstep;
    c00 = __builtin_amdgcn_wmma_f32_16x16x4_f32(false, a0, false, b0, (short)0, c00, false, false);
    c01 = __builtin_amdgcn_wmma_f32_16x16x4_f32(false, a0, false, b1, (short)0, c01, false, false);
    c10 = __builtin_amdgcn_wmma_f32_16x16x4_f32(false, a1, false, b0, (short)0, c10, false, false);
    c11 = __builtin_amdgcn_wmma_f32_16x16x4_f32(false, a1, false, b1, (short)0, c11, false, false);
  }
  int r0 = rowBase + hi * 8;                // C/D layout: lanes>=16 hold rows 8..15
  int r1 = r0 + 16;
  float* t0 = T + r0 * fo + col0;
  float* t1 = T + r1 * fo + col0;
  #pragma unroll
  for (int v = 0; v < 8; v++) {
    t0[0]  = c00[v];
    t0[16] = c01[v];
    t1[0]  = c10[v];
    t1[16] = c11[v];
    t0 += fo;
    t1 += fo;
  }
#else
  int r0 = rowBase + hi * 8;
  for (int rt = 0; rt < 2; rt++) {
    for (int v = 0; v < 8; v++) {
      const float* ar = A + (r0 + rt * 16 + v) * fi;
      float acc0 = 0.f, acc1 = 0.f;
      for (int k = 0; k < fi; k++) {
        acc0 += ar[k] * W[k * fo + col0];
        acc1 += ar[k] * W[k * fo + col1];
      }
      T[(r0 + rt * 16 + v) * fo + col0] = acc0;
      T[(r0 + rt * 16 + v) * fo + col1] = acc1;
    }
  }
#endif
}

// CSR gather aggregation fused with +bias, relu. One wave per node.
// Edge records fetched coalesced in chunks of 32, broadcast via shfl.
__global__ __launch_bounds__(256)
void k_agg(const float* __restrict__ t, float* __restrict__ out,
           const float* __restrict__ nself, const int* __restrict__ rowp,
           const int* __restrict__ rowend, const int* __restrict__ csrS,
           const float* __restrict__ csrN, const float* __restrict__ bias,
           int n, int fo) {
  int v = blockIdx.x * 8 + (threadIdx.x >> 5);
  if (v >= n) return;
  int lane = threadIdx.x & 31;
  int nf = fo >> 5;                         // features per lane (1..4)
  float acc[4];
  float sn = nself[v];
  const float* tv = t + v * fo;
  #pragma unroll 4
  for (int c = 0; c < nf; c++) acc[c] = sn * tv[lane + 32 * c];
  int beg = rowp[v], end = rowend[v];
  for (int j = beg; j < end; j += 32) {
    int jj = j + lane;
    int   sE = 0;
    float wE = 0.f;
    if (jj < end) { sE = csrS[jj]; wE = csrN[jj]; }   // coalesced chunk load
    int m = end - j;
    if (m > 32) m = 32;
    for (int q = 0; q < m; q++) {
      int   s  = __shfl(sE, q);
      float wv = __shfl(wE, q);
      const float* tr = t + s * fo;
      #pragma unroll 4
      for (int c = 0; c < nf; c++) acc[c] += wv * tr[lane + 32 * c];
    }
  }
  #pragma unroll 4
  for (int c = 0; c < nf; c++) {
    float r = acc[c] + bias[lane + 32 * c];
    out[v * fo + lane + 32 * c] = r > 0.f ? r : 0.f;
  }
}

// ---------------- pooling + MLP ----------------

__global__ void k_pool_init(unsigned* g, int sz) {
  int i = blockIdx.x * blockDim.x + threadIdx.x;
  if (i < sz) g[i] = 0u;                    // relu outputs >= 0, so 0 is -inf here
}

// max over nodes per graph; values are post-relu (>=0) so uint-bit max == float max
__global__ void k_pool(const float* __restrict__ h, const int* __restrict__ batch,
                       unsigned* g, int n) {
  int idx = blockIdx.x * blockDim.x + threadIdx.x;
  if (idx < n * 32) {
    int v = idx >> 5, f = idx & 31;
    atomicMax(&g[batch[v] * 32 + f], __float_as_uint(h[idx]));
  }
}

__global__ void k_mlp(const unsigned* __restrict__ g, const float* __restrict__ W5,
                      const float* __restrict__ b5, const float* __restrict__ W6,
                      const float* __restrict__ b6, float* __restrict__ out, int G) {
  __shared__ float sg[64 * 32];
  __shared__ float sg2[64 * 32];
  int tid = threadIdx.x;
  for (int i = tid; i < G * 32; i += blockDim.x) sg[i] = __uint_as_float(g[i]);
  __syncthreads();
  for (int idx = tid; idx < G * 32; idx += blockDim.x) {
    int i = idx >> 5, j = idx & 31;
    float acc = b5[j];
    #pragma unroll
    for (int k = 0; k < 32; k++) acc += sg[i * 32 + k] * W5[k * 32 + j];
    sg2[idx] = acc > 0.f ? acc : 0.f;
  }
  __syncthreads();
  for (int idx = tid; idx < G * 2; idx += blockDim.x) {
    int i = idx >> 1, j = idx & 1;
    float acc = b6[j];
    #pragma unroll
    for (int k = 0; k < 32; k++) acc += sg2[i * 32 + k] * W6[k * 2 + j];
    out[idx] = acc;
  }
}

// ---------------- host ----------------

extern "C" void kernel_launch(void* const* d_in, const int* in_sizes, int n_in,
                              void* d_out, int out_size, void* d_ws, size_t ws_size,
                              hipStream_t stream) {
  const float* x  = (const float*)d_in[0];
  const int*  ei  = (const int*)d_in[1];
  const float* ew = (const float*)d_in[2];
  const int*  bat = (const int*)d_in[3];
  const float *W1 = (const float*)d_in[4],  *b1 = (const float*)d_in[5];
  const float *W2 = (const float*)d_in[6],  *b2 = (const float*)d_in[7];
  const float *W3 = (const float*)d_in[8],  *b3 = (const float*)d_in[9];
  const float *W4 = (const float*)d_in[10], *b4 = (const float*)d_in[11];
  const float *W5 = (const float*)d_in[12], *b5 = (const float*)d_in[13];
  const float *W6 = (const float*)d_in[14], *b6 = (const float*)d_in[15];
  (void)n_in; (void)ws_size;

  const int N = in_sizes[0];     // x is [N,1]
  const int E = in_sizes[2];     // edge_weight is [E]
  const int G = out_size / 2;    // output [G,2]

  char* p = (char*)d_ws;
  auto carve = [&](size_t bytes) -> void* {
    void* r = p; p += (bytes + 255) & ~(size_t)255; return r;
  };
  float*    hA   = (float*)carve((size_t)N * 128 * 4);   // ping
  float*    hB   = (float*)carve((size_t)N * 128 * 4);   // pong (GEMM output t)
  float*    dis  = (float*)carve((size_t)N * 4);         // deg, then rsqrt(deg)
  float*    nsf  = (float*)carve((size_t)N * 4);         // self-loop norm
  int*      cnt  = (int*)carve((size_t)N * 4);
  int*      rowp = (int*)carve((size_t)N * 4);
  int*      cur  = (int*)carve((size_t)N * 4);
  int*      bsum = (int*)carve(4096);
  int*      boff = (int*)carve(4096);
  int*      csrS = (int*)carve((size_t)E * 4);
  float*    csrN = (float*)carve((size_t)E * 4);
  unsigned* g    = (unsigned*)carve((size_t)G * 32 * 4);

  const int nb = (N + 511) / 512;

  // graph normalization + CSR build (once, reused by all 4 layers)
  k_init   <<<(N + 255) / 256, 256, 0, stream>>>(cnt, dis, N);
  k_count  <<<(E + 255) / 256, 256, 0, stream>>>(ei, ew, cnt, dis, E);
  k_dis    <<<(N + 255) / 256, 256, 0, stream>>>(dis, nsf, N);
  k_scan1  <<<nb, 512, 0, stream>>>(cnt, rowp, bsum, N);
  k_scan2  <<<1, 512, 0, stream>>>(bsum, boff, nb);
  k_scan3  <<<(N + 255) / 256, 256, 0, stream>>>(rowp, boff, cur, N);
  k_scatter<<<(E + 255) / 256, 256, 0, stream>>>(ei, ew, dis, cur, csrS, csrN, E);

  // 8 waves/block, one 32x32 tile per wave -> 256 rows per block
  const int gemmBlk = (N + 255) / 256;

  // layer 1: x[N,1] @ W1[1,128] -> aggregate(+b1, relu)
  k_gemm1<<<(N * 32 + 255) / 256, 256, 0, stream>>>(x, W1, hB, N);
  k_agg<<<(N + 7) / 8, 256, 0, stream>>>(hB, hA, nsf, rowp, cur, csrS, csrN, b1, N, 128);
  // layer 2: 128 -> 96
  k_gemm<<<dim3(gemmBlk, 96 / 32), 256, 0, stream>>>(hA, W2, hB, N, 128, 96);
  k_agg<<<(N + 7) / 8, 256, 0, stream>>>(hB, hA, nsf, rowp, cur, csrS, csrN, b2, N, 96);
  // layer 3: 96 -> 64
  k_gemm<<<dim3(gemmBlk, 64 / 32), 256, 0, stream>>>(hA, W3, hB, N, 96, 64);
  k_agg<<<(N + 7) / 8, 256, 0, stream>>>(hB, hA, nsf, rowp, cur, csrS, csrN, b3, N, 64);
  // layer 4: 64 -> 32
  k_gemm<<<dim3(gemmBlk, 32 / 32), 256, 0, stream>>>(hA, W4, hB, N, 64, 32);
  k_agg<<<(N + 7) / 8, 256, 0, stream>>>(hB, hA, nsf, rowp, cur, csrS, csrN, b4, N, 32);

  // global max pool + MLP head
  k_pool_init<<<(G * 32 + 255) / 256, 256, 0, stream>>>(g, G * 32);
  k_pool<<<(N * 32 + 255) / 256, 256, 0, stream>>>(hA, bat, g, N);
  k_mlp<<<1, 1024, 0, stream>>>(g, W5, b5, W6, b6, (float*)d_out, G);
}